// RecurrentGCN_29841432772746
// MI455X (gfx1250) — compile-verified
//
#include <hip/hip_runtime.h>
#include <hip/hip_bf16.h>
#include <math.h>

typedef __attribute__((ext_vector_type(2))) float v2f;
typedef __attribute__((ext_vector_type(8))) float v8f;

#define HID  100
#define FEAT 128
#define NT1  13    // stage-1 N tiles (208 padded cols = [cz|ch])
#define KS1  32    // stage-1 K steps (128 / 4)
#define NT2  7     // stage-2 N tiles (112 padded cols)
#define KS2  25    // stage-2 K steps (100 / 4)

// ---------------------------------------------------------------- weight packs
__device__ __forceinline__ float src_w1(const float* Wcz, const float* Wch,
                                        int k, int col) {
  if (col < HID)      return Wcz[k * HID + col];
  if (col < 2 * HID)  return Wch[k * HID + (col - HID)];
  return 0.f;
}

__global__ void k_pack1(const float* __restrict__ Wcz,
                        const float* __restrict__ Wch,
                        float* __restrict__ pack1) {
  int idx = blockIdx.x * 256 + threadIdx.x;
  if (idx >= NT1 * KS1 * 32) return;
  int l = idx & 31, sl = idx >> 5;
  int s = sl % KS1, t = sl / KS1;
  int k0  = s * 4 + ((l >> 4) << 1);
  int col = t * 16 + (l & 15);
  float2 v;
  v.x = src_w1(Wcz, Wch, k0,     col);
  v.y = src_w1(Wcz, Wch, k0 + 1, col);
  ((float2*)pack1)[idx] = v;
}

__global__ void k_pack2(const float* __restrict__ Wlz,
                        const float* __restrict__ Wlh,
                        float* __restrict__ pack2) {
  const int per = NT2 * KS2 * 32;
  int idx = blockIdx.x * 256 + threadIdx.x;
  if (idx >= 2 * per) return;
  int g = idx / per, rem = idx % per;
  int l = rem & 31, sl = rem >> 5;
  int s = sl % KS2, t = sl / KS2;
  const float* W = g ? Wlh : Wlz;          // (200,100) row-major, top half used
  int k0  = s * 4 + ((l >> 4) << 1);
  int col = t * 16 + (l & 15);
  float2 v;
  v.x = (k0     < HID && col < HID) ? W[(k0)     * HID + col] : 0.f;
  v.y = (k0 + 1 < HID && col < HID) ? W[(k0 + 1) * HID + col] : 0.f;
  ((float2*)pack2)[idx] = v;
}

// ---------------------------------------------------------------- degree / norm
__global__ void k_deg_init(float* __restrict__ deg, int N) {
  int i = blockIdx.x * 256 + threadIdx.x;
  if (i < N) deg[i] = 1.0f;                // self-loop weight
}

__global__ void k_deg_edge(const long long* __restrict__ ei,
                           const float* __restrict__ w,
                           float* __restrict__ deg, int E) {
  int e = blockIdx.x * 256 + threadIdx.x;
  if (e < E) unsafeAtomicAdd(&deg[ei[E + e]], w[e]);   // dst-segment sum
}

__global__ void k_dinv(const float* __restrict__ deg,
                       float* __restrict__ dinv, int N) {
  int i = blockIdx.x * 256 + threadIdx.x;
  if (i < N) { float d = deg[i]; dinv[i] = (d > 0.f) ? rsqrtf(d) : 0.f; }
}

// agg = dinv^2 * x  (self-loop term), then atomic edge scatter
__global__ void k_agg_init(const float* __restrict__ x,
                           const float* __restrict__ dinv,
                           float* __restrict__ agg, int N) {
  int i = blockIdx.x * 256 + threadIdx.x;
  if (i < N * FEAT) {
    int node = i >> 7;
    float di = dinv[node];
    agg[i] = di * di * x[i];
  }
}

__global__ void k_agg_edge(const long long* __restrict__ ei,
                           const float* __restrict__ w,
                           const float* __restrict__ x,
                           const float* __restrict__ dinv,
                           float* __restrict__ agg, int E) {
  int wid  = (int)((blockIdx.x * blockDim.x + threadIdx.x) >> 5);
  int lane = threadIdx.x & 31;
  int nw   = (int)((gridDim.x * blockDim.x) >> 5);
  for (int e = wid; e < E; e += nw) {
    long long s = ei[e], d = ei[E + e];
    float norm = dinv[s] * w[e] * dinv[d];
    float4 v = ((const float4*)(x + s * FEAT))[lane];   // coalesced 512B/wave
    float* ar = agg + d * FEAT + lane * 4;
    unsafeAtomicAdd(ar + 0, norm * v.x);
    unsafeAtomicAdd(ar + 1, norm * v.y);
    unsafeAtomicAdd(ar + 2, norm * v.z);
    unsafeAtomicAdd(ar + 3, norm * v.w);
  }
}

// ---------------------------------------------------------------- fused WMMA GEMMs
// per block: 32 nodes.  stage1: agg(32x128) @ [Wcz|Wch](128x208) -> S1 (+bias)
//            stage2: S1[:,0:100] @ Wlz_top, S1[:,100:200] @ Wlh_top -> S2
//            epilogue: sigmoid/tanh/relu + dot(Wout) -> out
__global__ __launch_bounds__(256) void k_fused(
    const float* __restrict__ agg,
    const float* __restrict__ pack1, const float* __restrict__ pack2,
    const float* __restrict__ bcz, const float* __restrict__ bch,
    const float* __restrict__ blz, const float* __restrict__ blh,
    const float* __restrict__ Wout, const float* __restrict__ bout,
    float* __restrict__ out, int N) {
  __shared__ __align__(16) float ldsU[32 * 228];   // A tile (32x132), reused as S2 (32x228)
  __shared__ __align__(16) float ldsS1[32 * 212];  // stage-1 result, padded stride
  __shared__ float red[32][8];

  const int tid  = threadIdx.x;
  const int wave = tid >> 5, lane = tid & 31;
  const int nb   = blockIdx.x * 32;
  const int lrow  = lane & 15;
  const int khalf = (lane >> 4) << 1;              // 0 or 2 (K-half select)
  const int rbase = (lane >> 4) << 3;              // C/D row base per lane half

  // phase 0: cooperative coalesced A-tile load (stride 132 -> conflict-free frags)
  for (int idx = tid; idx < 32 * FEAT; idx += 256) {
    int row = idx >> 7, k = idx & 127;
    int node = nb + row;
    ldsU[row * 132 + k] = (node < N) ? agg[(long long)node * FEAT + k] : 0.f;
  }
  __syncthreads();

  // phase 1: stage-1 GEMM (M=32, N=208, K=128) via V_WMMA_F32_16X16X4_F32
  for (int t = wave; t < 2 * NT1; t += 8) {
    int mt = t / NT1, nt = t % NT1;
    v8f acc = {};
    const v2f* bp = (const v2f*)pack1 + (nt * KS1) * 32 + lane;
    for (int s = 0; s < KS1; ++s) {
      v2f a = *(const v2f*)&ldsU[(mt * 16 + lrow) * 132 + s * 4 + khalf];
      v2f b = bp[s * 32];
      acc = __builtin_amdgcn_wmma_f32_16x16x4_f32(false, a, false, b,
                                                  (short)0, acc, false, false);
    }
    int colb = nt * 16 + lrow;
    float bias = (colb < HID) ? bcz[colb]
               : (colb < 2 * HID) ? bch[colb - HID] : 0.f;
    for (int r = 0; r < 8; ++r)
      ldsS1[(mt * 16 + rbase + r) * 212 + colb] = acc[r] + bias;
  }
  __syncthreads();

  // phase 2: stage-2 GEMMs (g=0: Z-gate, g=1: H-candidate), K=100, N=112
  for (int t = wave; t < 2 * 2 * NT2; t += 8) {
    int g  = t / (2 * NT2);
    int rr = t % (2 * NT2);
    int mt = rr / NT2, nt = rr % NT2;
    v8f acc = {};
    const v2f* bp = (const v2f*)pack2 + ((g * NT2 + nt) * KS2) * 32 + lane;
    for (int s = 0; s < KS2; ++s) {
      v2f a = *(const v2f*)&ldsS1[(mt * 16 + lrow) * 212 + g * HID + s * 4 + khalf];
      v2f b = bp[s * 32];
      acc = __builtin_amdgcn_wmma_f32_16x16x4_f32(false, a, false, b,
                                                  (short)0, acc, false, false);
    }
    int colb = g * 112 + nt * 16 + lrow;
    for (int r = 0; r < 8; ++r)
      ldsU[(mt * 16 + rbase + r) * 228 + colb] = acc[r];
  }
  __syncthreads();

  // phase 3: gates + output projection (8 threads per node)
  int node = tid >> 3, part = tid & 7;
  float sum = 0.f;
  for (int c = part; c < HID; c += 8) {
    float zv = ldsU[node * 228 + c] + blz[c];
    float z  = 1.f / (1.f + __expf(-zv));
    float hv = ldsU[node * 228 + 112 + c] + blh[c];
    float h  = (1.f - z) * tanhf(hv);     // Hn = (1-Z)*Ht  (H0 == 0)
    sum += fmaxf(h, 0.f) * Wout[c];
  }
  red[node][part] = sum;
  __syncthreads();
  if (part == 0 && (nb + node) < N) {
    float tot = 0.f;
    for (int p = 0; p < 8; ++p) tot += red[node][p];
    out[nb + node] = tot + bout[0];
  }
}

// ---------------------------------------------------------------- launcher
extern "C" void kernel_launch(void* const* d_in, const int* in_sizes, int n_in,
                              void* d_out, int out_size, void* d_ws, size_t ws_size,
                              hipStream_t stream) {
  const float*     x    = (const float*)d_in[0];
  const long long* ei   = (const long long*)d_in[1];
  const float*     ew   = (const float*)d_in[2];
  const float*     Wcz  = (const float*)d_in[3];
  const float*     bcz  = (const float*)d_in[4];
  // d_in[5..6] = Wcr/bcr : dead (R only multiplies H0 == 0)
  const float*     Wch  = (const float*)d_in[7];
  const float*     bch  = (const float*)d_in[8];
  const float*     Wlz  = (const float*)d_in[9];
  const float*     blz  = (const float*)d_in[10];
  // d_in[11..12] = Wlr/blr : dead
  const float*     Wlh  = (const float*)d_in[13];
  const float*     blh  = (const float*)d_in[14];
  const float*     Wout = (const float*)d_in[15];
  const float*     bout = (const float*)d_in[16];
  float* out = (float*)d_out;

  const int N = in_sizes[0] / FEAT;
  const int E = in_sizes[2];

  float* ws    = (float*)d_ws;
  float* agg   = ws;                                  // N*128 floats
  float* deg   = agg  + (size_t)N * FEAT;             // N
  float* dinv  = deg  + N;                            // N
  float* pack1 = dinv + N;                            // 13*32*32*2
  float* pack2 = pack1 + NT1 * KS1 * 32 * 2;          // 2*7*25*32*2

  k_pack1<<<(NT1 * KS1 * 32 + 255) / 256, 256, 0, stream>>>(Wcz, Wch, pack1);
  k_pack2<<<(2 * NT2 * KS2 * 32 + 255) / 256, 256, 0, stream>>>(Wlz, Wlh, pack2);

  k_deg_init<<<(N + 255) / 256, 256, 0, stream>>>(deg, N);
  k_deg_edge<<<(E + 255) / 256, 256, 0, stream>>>(ei, ew, deg, E);
  k_dinv<<<(N + 255) / 256, 256, 0, stream>>>(deg, dinv, N);

  k_agg_init<<<(N * FEAT + 255) / 256, 256, 0, stream>>>(x, dinv, agg, N);
  k_agg_edge<<<(E + 7) / 8, 256, 0, stream>>>(ei, ew, x, dinv, agg, E);

  k_fused<<<(N + 31) / 32, 256, 0, stream>>>(agg, pack1, pack2,
                                             bcz, bch, blz, blh,
                                             Wout, bout, out, N);
}